// ConcatJointNet_12841952215581
// MI455X (gfx1250) — compile-verified
//
#include <hip/hip_runtime.h>
#include <hip/hip_bf16.h>

// Problem dims (fixed by the reference's setup_inputs).
#define T_DIM 256
#define B_DIM 8
#define U_DIM 128
#define DE    512
#define DD    512
#define H_DIM 512

typedef __attribute__((ext_vector_type(2))) float v2f;
typedef __attribute__((ext_vector_type(4))) float v4f;
typedef __attribute__((ext_vector_type(8))) float v8f;

// Native CDNA5 tanh (TRANS op). Prefer the clang builtin (compiler handles
// TRANS-op hazards); otherwise emit v_tanh_f32 via inline asm with a trailing
// v_nop per the ISA hazard rule (a trans op needs one independent op or nop
// before its result is used or its sources overwritten). The 4 back-to-back tanh ops are
// mutually independent, so only the last one needs the explicit v_nop.
__device__ __forceinline__ v4f tanh4(v4f s) {
#if __has_builtin(__builtin_amdgcn_tanhf)
  v4f r;
  r.x = __builtin_amdgcn_tanhf(s.x);
  r.y = __builtin_amdgcn_tanhf(s.y);
  r.z = __builtin_amdgcn_tanhf(s.z);
  r.w = __builtin_amdgcn_tanhf(s.w);
  return r;
#else
#warning "CDNA5: __builtin_amdgcn_tanhf not available; using inline asm v_tanh_f32"
  float r0, r1, r2, r3;
  asm volatile("v_tanh_f32 %0, %4\n\t"
               "v_tanh_f32 %1, %5\n\t"
               "v_tanh_f32 %2, %6\n\t"
               "v_tanh_f32 %3, %7\n\t"
               "v_nop"
               : "=&v"(r0), "=&v"(r1), "=&v"(r2), "=&v"(r3)
               : "v"(s.x), "v"(s.y), "v"(s.z), "v"(s.w));
  v4f r = {r0, r1, r2, r3};
  return r;
#endif
}

// ---------------------------------------------------------------------------
// Kernel 1: fused GEMM via V_WMMA_F32_16X16X4_F32 (fp32 in, fp32 acc).
//   rows [0, B*T)        : enc_proj[b*T+t][h] = sum_d enc[t][b][d] * W[d][h]
//   rows [B*T, B*T+B*U)  : dec_proj[b*U+u][h] = sum_d dec[b][u][d] * W[De+d][h]
// One wave (32 lanes) computes one 16x16 tile of the output; K loop steps 4.
//
// ISA VGPR layouts (CDNA5 ISA):
//   A (16x4):  lanes 0-15 -> M=lane, {v0,v1}={K=0,K=1}; lanes 16-31 -> {K=2,K=3}
//   B (4x16):  lanes 0-15 -> N=lane, {v0,v1}={K=0,K=1}; lanes 16-31 -> {K=2,K=3}
//   C/D (16x16): v[r], lanes 0-15 -> M=r, lanes 16-31 -> M=r+8, N=lane&15
// ---------------------------------------------------------------------------
__global__ void __launch_bounds__(32)
joint_proj_wmma(const float* __restrict__ enc,    // (T, B, De)
                const float* __restrict__ dec,    // (B, U, Dd)
                const float* __restrict__ W,      // (De+Dd, H)
                float* __restrict__ enc_proj,     // (B*T, H)
                float* __restrict__ dec_proj)     // (B*U, H)
{
  const int lane = threadIdx.x;                 // 0..31, EXEC all-ones
  const int n0   = blockIdx.x * 16;             // H tile
  const int tileM = blockIdx.y;
  const int m    = lane & 15;
  const int kh   = (lane >> 4) << 1;            // 0 or 2 (K sub-offset per half-wave)

  const int encTiles = (B_DIM * T_DIM) / 16;    // 128
  const bool is_dec  = tileM >= encTiles;

  const float* arow;
  int kbase;
  float* orow;
  if (is_dec) {
    const int r = (tileM - encTiles) * 16 + m;  // r = b*U + u, dec is (B,U,Dd) contiguous
    arow  = dec + (size_t)r * DD;
    kbase = DE;
    orow  = dec_proj + (size_t)(tileM - encTiles) * 16 * H_DIM;
  } else {
    const int r = tileM * 16 + m;               // logical row r -> (b = r/T, t = r%T)
    const int b = r / T_DIM;
    const int t = r % T_DIM;
    arow  = enc + ((size_t)t * B_DIM + b) * DE; // enc[t][b][:]
    kbase = 0;
    orow  = enc_proj + (size_t)tileM * 16 * H_DIM;
  }

  // Per-lane B column pointer: W[kbase + k][n0 + (lane&15)]
  const float* wcol = W + (size_t)kbase * H_DIM + (n0 + (lane & 15));

  v8f acc = {0.f, 0.f, 0.f, 0.f, 0.f, 0.f, 0.f, 0.f};

#pragma unroll 4
  for (int k0 = 0; k0 < DE; k0 += 4) {
    v2f a;
    a.x = arow[k0 + kh];
    a.y = arow[k0 + kh + 1];
    v2f bm;
    bm.x = wcol[(size_t)(k0 + kh)     * H_DIM];
    bm.y = wcol[(size_t)(k0 + kh + 1) * H_DIM];
    // 8 args: (neg_a, A, neg_b, B, c_mod, C, reuse_a, reuse_b)
    acc = __builtin_amdgcn_wmma_f32_16x16x4_f32(
        false, a, false, bm, (short)0, acc, false, false);
  }

  const int nOut  = lane & 15;
  const int mBase = (lane >> 4) * 8;
#pragma unroll
  for (int r = 0; r < 8; ++r) {
    orow[(size_t)(mBase + r) * H_DIM + n0 + nOut] = acc[r];
  }
}

// ---------------------------------------------------------------------------
// Kernel 2: h_joint[b][t][u][h] = tanh(enc_proj[b*T+t][h] + dec_proj[b*U+u][h] + bias[h])
// Pure streaming-store kernel (512 MB out, ~22 us floor at 23.3 TB/s).
// All inputs (6 MB proj + bias) sit in the 192 MB L2, so reads are cheap.
// One block per (b,t); 256 threads cover H/4 float4 lanes x 2 u-phases.
// Non-temporal b128 stores: output >> L2, don't cache it.
// ---------------------------------------------------------------------------
__global__ void __launch_bounds__(256)
joint_tanh(const float* __restrict__ enc_proj,   // (B*T, H)
           const float* __restrict__ dec_proj,   // (B*U, H)
           const float* __restrict__ bias,       // (H,)
           float* __restrict__ out)              // (B, T, U, H)
{
  const int bt = blockIdx.x;                     // b*T + t
  const int b  = bt / T_DIM;

  const float* ep = enc_proj + (size_t)bt * H_DIM;
  const float* dp = dec_proj + (size_t)b * U_DIM * H_DIM;
  float*       op = out      + (size_t)bt * U_DIM * H_DIM;

  const int hv = (threadIdx.x & 127) << 2;       // float4-aligned h offset
  const int u0 = threadIdx.x >> 7;               // 0 or 1

  const v4f e  = *(const v4f*)(ep + hv);
  const v4f bb = *(const v4f*)(bias + hv);
  const v4f eb = e + bb;                         // fold bias once per thread

  for (int u = u0; u < U_DIM; u += 2) {
    const v4f d = *(const v4f*)(dp + (size_t)u * H_DIM + hv);
    const v4f r = tanh4(eb + d);
    __builtin_nontemporal_store(r, (v4f*)(op + (size_t)u * H_DIM + hv));
  }
}

// ---------------------------------------------------------------------------
// Kernel 3: fake_src_lengths[b] = sum_t (int(mask[b][t]) != padding_idx)
// (DOWNSAMPLE == 1 -> ceil is identity). Output dtype int32.
// ---------------------------------------------------------------------------
__global__ void __launch_bounds__(256)
src_lengths(const unsigned char* __restrict__ mask,   // (B, T) bool
            const int* __restrict__ pad_idx,          // scalar
            int* __restrict__ out_len)                // (B,)
{
  __shared__ int red[256];
  const int b = blockIdx.x;
  const int t = threadIdx.x;
  const int pidx = pad_idx[0];
  red[t] = ((int)mask[b * T_DIM + t] != pidx) ? 1 : 0;
  __syncthreads();
#pragma unroll
  for (int s = 128; s > 0; s >>= 1) {
    if (t < s) red[t] += red[t + s];
    __syncthreads();
  }
  if (t == 0) out_len[b] = red[0];
}

extern "C" void kernel_launch(void* const* d_in, const int* in_sizes, int n_in,
                              void* d_out, int out_size, void* d_ws, size_t ws_size,
                              hipStream_t stream) {
  (void)in_sizes; (void)n_in; (void)out_size; (void)ws_size;

  const float*         enc     = (const float*)d_in[0];          // (T,B,De)
  const float*         dec     = (const float*)d_in[1];          // (B,U,Dd)
  const unsigned char* mask    = (const unsigned char*)d_in[2];  // (B,T) bool
  const float*         W       = (const float*)d_in[3];          // (De+Dd,H)
  const float*         bias    = (const float*)d_in[4];          // (H,)
  const int*           pad_idx = (const int*)d_in[5];            // scalar

  float* out      = (float*)d_out;
  float* enc_proj = (float*)d_ws;                                        // 4 MB
  float* dec_proj = enc_proj + (size_t)B_DIM * T_DIM * H_DIM;            // 2 MB

  // 1) WMMA GEMMs: (B*T + B*U)/16 = 192 row-tiles, H/16 = 32 col-tiles.
  dim3 g1(H_DIM / 16, (B_DIM * T_DIM + B_DIM * U_DIM) / 16);
  joint_proj_wmma<<<g1, 32, 0, stream>>>(enc, dec, W, enc_proj, dec_proj);

  // 2) Broadcast + tanh + streaming store (the 512 MB bandwidth-bound part).
  joint_tanh<<<dim3(B_DIM * T_DIM), 256, 0, stream>>>(enc_proj, dec_proj, bias, out);

  // 3) Lengths, appended after h_joint in d_out.
  int* out_len = (int*)(out + (size_t)B_DIM * T_DIM * U_DIM * H_DIM);
  src_lengths<<<dim3(B_DIM), 256, 0, stream>>>(mask, pad_idx, out_len);
}